// cgss_14104672600182
// MI455X (gfx1250) — compile-verified
//
#include <hip/hip_runtime.h>
#include <hip/hip_bf16.h>

// Fused Gaussian-blur(C-axis) + LIF scan(T-axis) for (B,1,C,T) fp32.
// Blur as banded matmul on the WMMA pipe: per 16ch x 16t tile,
// mean = K(16x28) * X(28x16) via 7x V_WMMA_F32_16X16X4_F32.
// Staging: GLOBAL_LOAD_ASYNC_TO_LDS_B128 double-buffered in LDS (ASYNCcnt),
// falling back to register prefetch if the async builtin is unavailable.
// Memory-bound (134 MB @ 23.3 TB/s ~ 6 us): single fused pass, no scratch.

typedef __attribute__((ext_vector_type(2))) float v2f;
typedef __attribute__((ext_vector_type(4))) float v4f;
typedef __attribute__((ext_vector_type(8))) float v8f;
typedef __attribute__((ext_vector_type(4))) int   v4i;

#define GAS __attribute__((address_space(1)))
#define LAS __attribute__((address_space(3)))

#if defined(__has_builtin)
#if __has_builtin(__builtin_amdgcn_global_load_async_to_lds_b128)
#define HAVE_ASYNC 1
#endif
#endif
#ifndef HAVE_ASYNC
#define HAVE_ASYNC 0
#endif

#define B_    32
#define C_    512
#define T_    1024
#define KS    13
#define KH    6          // KSIZE/2
#define THR_  0.5f
#define BETA_ 0.95f
#define NCH   28         // 16 output channels + 2*6 halo
#define NG    7          // ceil(28/4) WMMA k-chunks
#define SDP   20         // sd pitch in floats: 16B-aligned rows + conflict-free scan

#if HAVE_ASYNC
#define NBUF  2
// Copy 16B global -> LDS via the async DMA path (no destination VGPRs).
// Builtin prototype: (v4i AS1* src, LDS dst, imm offset -> both sides, imm cpol).
template <int OFF>
static __device__ __forceinline__ void async_ld16(const float* g, float* l) {
  __builtin_amdgcn_global_load_async_to_lds_b128((GAS v4i*)g, (LAS v4i*)l, OFF, 0);
}
#else
#define NBUF  1
#endif

__global__ __launch_bounds__(32)
void snn_gauss_lif_fused(const float* __restrict__ in,
                         const float* __restrict__ wp,
                         float* __restrict__ out) {
  __shared__ float sk[KS];             // gaussian taps (unnormalized)
  __shared__ float sx[NBUF][NCH * 16]; // staged input tile(s): 28 rows x 16 t
  __shared__ float sd[16 * SDP];       // drive/spike tile, pitch 20

  const int lane = threadIdx.x;     // wave32
  const int gl   = lane >> 4;       // lane half-group (0: lanes 0-15, 1: 16-31)
  const int n    = lane & 15;       // time-within-tile / M index
  const int c0   = blockIdx.x * 16; // channel tile base
  const int b    = blockIdx.y;

  // ---- Gaussian taps from scalar w: sigma = 5.5 + clip(w,1,10) ----
  const float w     = wp[0];
  const float sigma = 0.5f * (1.0f + 10.0f) + fminf(fmaxf(w, 1.0f), 10.0f);
  if (lane < KS) {
    const float r = (float)(lane - KH) / sigma;
    sk[lane] = __expf(-0.5f * r * r);
  }
  __syncthreads();
  float ksum = 0.0f;
#pragma unroll
  for (int j = 0; j < KS; ++j) ksum += sk[j];
  const float kinv = 1.0f / ksum;

  // ---- A fragments: banded kernel matrix, constant over all time tiles ----
  // ISA layout (16x4 f32 A): vgpr j, lanes 0-15 hold K=j, lanes 16-31 K=j+2;
  // M = lane&15.  A[m][kk] couples out-ch (c0+m) with window row (4g+kk):
  // tap = 4g+kk-m, zero outside [0,13).
  v2f afrag[NG];
#pragma unroll
  for (int g = 0; g < NG; ++g) {
#pragma unroll
    for (int j = 0; j < 2; ++j) {
      const int kk = j + 2 * gl;
      const int jt = 4 * g + kk - n;
      afrag[g][j] = (jt >= 0 && jt < KS) ? sk[jt] * kinv : 0.0f;
    }
  }

  const float* __restrict__ inb  = in  + (size_t)b * C_ * T_;
  float* __restrict__       outb = out + (size_t)b * C_ * T_;

  // ---- Row-per-lane staging: lane l (<28) owns window row l (= ch c0-6+l).
  // Halo rows outside [0,C) are zero-filled ONCE and never overwritten.
  const int  srow   = (lane < NCH) ? lane : (NCH - 1);
  const int  ci     = c0 - KH + srow;
  const int  cic    = (ci < 0) ? 0 : ((ci > C_ - 1) ? (C_ - 1) : ci);
  const bool rvalid = (lane < NCH) && (ci >= 0) && (ci < C_);
  const float* __restrict__ rowbase = inb + (size_t)cic * T_;

  if (lane < NCH && !rvalid) {
#pragma unroll
    for (int ib = 0; ib < NBUF; ++ib) {
      v4f* sxr = (v4f*)&sx[ib][lane * 16];
#pragma unroll
      for (int q = 0; q < 4; ++q) sxr[q] = (v4f)0.0f;
    }
  }
  __syncthreads();

  float mem = 0.0f;  // LIF state; meaningful in lanes 0..15 (one channel each)

#if HAVE_ASYNC
  // prologue: async-fill buffer 0 with tile t0 = 0 (no VGPR staging at all)
  if (rvalid) {
    float* lrow = &sx[0][lane * 16];
    async_ld16< 0>(rowbase, lrow);
    async_ld16<16>(rowbase, lrow);
    async_ld16<32>(rowbase, lrow);
    async_ld16<48>(rowbase, lrow);
  }
#else
  // prologue: register prefetch of tile t0 = 0 (named v4f scalars: no spill)
  const v4f* __restrict__ rowv = (const v4f*)rowbase;
  v4f pr0 = rowv[0], pr1 = rowv[1], pr2 = rowv[2], pr3 = rowv[3];
#endif

  for (int t0 = 0; t0 < T_; t0 += 16) {
#if HAVE_ASYNC
    const int ib = (t0 >> 4) & 1;
    // current buffer's async fill complete + visible to all lanes
    asm volatile("s_wait_asynccnt 0" ::: "memory");
    __syncthreads();
    // issue async fill of the other buffer for the next tile; latency hidden
    // behind the whole WMMA + scan + store body. Prior reads of that buffer
    // were drained by the previous iteration's end barrier.
    {
      const int t1 = (t0 + 16 < T_) ? (t0 + 16) : t0;
      if (rvalid) {
        const float* grow = rowbase + t1;
        float*       lrow = &sx[ib ^ 1][lane * 16];
        async_ld16< 0>(grow, lrow);
        async_ld16<16>(grow, lrow);
        async_ld16<32>(grow, lrow);
        async_ld16<48>(grow, lrow);
      }
    }
#else
    const int ib = 0;
    // commit staged tile, then rotate-in-place prefetch of the next tile
    if (rvalid) {
      v4f* sxr = (v4f*)&sx[0][lane * 16];
      sxr[0] = pr0; sxr[1] = pr1; sxr[2] = pr2; sxr[3] = pr3;
    }
    __syncthreads();
    {
      const int t1 = (t0 + 16 < T_) ? (t0 + 16) : t0;
      const int tb = t1 >> 2;
      pr0 = rowv[tb]; pr1 = rowv[tb + 1]; pr2 = rowv[tb + 2]; pr3 = rowv[tb + 3];
    }
#endif
    const float* xs = &sx[ib][0];

    // phase 2: preload all B fragments, then chain 7 WMMAs on the acc.
    // B layout mirror of A: vgpr j, lane-group gl -> K = j+2*gl, N = lane&15.
    v2f bf[NG];
#pragma unroll
    for (int g = 0; g < NG; ++g) {
      bf[g][0] = xs[(4 * g + 0 + 2 * gl) * 16 + n];
      bf[g][1] = xs[(4 * g + 1 + 2 * gl) * 16 + n];
    }
    v8f acc = {};
#pragma unroll
    for (int g = 0; g < NG; ++g) {
      acc = __builtin_amdgcn_wmma_f32_16x16x4_f32(
          /*neg_a=*/false, afrag[g], /*neg_b=*/false, bf[g],
          /*c_mod=*/(short)0, acc, /*reuse_a=*/false, /*reuse_b=*/false);
    }

    // phase 3: drive = x - mean; C/D layout: vgpr v -> M = v + 8*gl, N = lane&15
#pragma unroll
    for (int v = 0; v < 8; ++v) {
      const int m = v + 8 * gl;
      sd[m * SDP + n] = xs[(KH + m) * 16 + n] - acc[v];
    }
    __syncthreads();

    // phase 4: LIF scan, lanes 0..15 own one channel each, 16 serial steps
    if (lane < 16) {
#pragma unroll
      for (int tt = 0; tt < 16; ++tt) {
        const float xt    = sd[lane * SDP + tt];
        const float reset = (mem > THR_) ? 1.0f : 0.0f;
        mem = BETA_ * mem + xt - reset * THR_;
        sd[lane * SDP + tt] = (mem > THR_) ? 1.0f : 0.0f;
      }
    }
    __syncthreads();

    // phase 5: store spike tile as b128 (8 x 64B contiguous runs per instr)
#pragma unroll
    for (int h = 0; h < 2; ++h) {
      const int i   = lane + 32 * h;   // 0..63 over 16 rows x 4 qwords
      const int row = i >> 2;
      const int qw  = i & 3;
      const v4f v = *(const v4f*)&sd[row * SDP + qw * 4];
      *(v4f*)&outb[(size_t)(c0 + row) * T_ + t0 + qw * 4] = v;
    }
    __syncthreads();
  }
}

extern "C" void kernel_launch(void* const* d_in, const int* in_sizes, int n_in,
                              void* d_out, int out_size, void* d_ws, size_t ws_size,
                              hipStream_t stream) {
  (void)in_sizes; (void)n_in; (void)d_ws; (void)ws_size; (void)out_size;
  const float* inp = (const float*)d_in[0];  // (B,1,C,T) fp32
  const float* w   = (const float*)d_in[1];  // scalar fp32
  float* out       = (float*)d_out;          // (B,1,C,T) fp32

  dim3 grid(C_ / 16, B_);   // 32 x 32 = 1024 single-wave blocks
  dim3 block(32);
  snn_gauss_lif_fused<<<grid, block, 0, stream>>>(inp, w, out);
}